// LenPredictorAttention_86809878987111
// MI455X (gfx1250) — compile-verified
//
#include <hip/hip_runtime.h>

// Problem constants (from the reference)
#define B_DIM   16
#define ENC_T   512
#define ENC_DIM 512
#define DEC_T   2048

typedef __attribute__((ext_vector_type(2))) float v2f;
typedef __attribute__((ext_vector_type(4))) float v4f;
typedef __attribute__((ext_vector_type(8))) float v8f;

// ---------------------------------------------------------------------------
// Kernel 1: per-batch cumulative durations + decoder-step -> phoneme map.
//   cum[b][e]    = inclusive cumsum of integer durations
//   e_of_t[b][t] = phoneme index covering decoder step t, or -1 if t >= total
// One block per batch row; LDS two-level scan then per-t binary search.
// ---------------------------------------------------------------------------
__global__ __launch_bounds__(256)
void build_alignment_map(const float* __restrict__ enc_len,
                         int* __restrict__ cum,
                         int* __restrict__ e_of_t)
{
    const int b   = blockIdx.x;
    const int tid = threadIdx.x;

    __shared__ int s_cum[ENC_T];
    __shared__ int s_tot[16];

    for (int i = tid; i < ENC_T; i += 256)
        s_cum[i] = (int)enc_len[b * ENC_T + i];
    __syncthreads();

    // inclusive scan: 16 chunks of 32 (serial per chunk), then chunk offsets
    if (tid < 16) {
        int acc = 0;
        const int base = tid * 32;
        for (int j = 0; j < 32; ++j) { acc += s_cum[base + j]; s_cum[base + j] = acc; }
        s_tot[tid] = acc;
    }
    __syncthreads();
    if (tid == 0) {
        int acc = 0;
        for (int i = 0; i < 16; ++i) { int v = s_tot[i]; s_tot[i] = acc; acc += v; }
    }
    __syncthreads();
    {
        const int i0  = tid * 2;
        const int off = s_tot[i0 >> 5];
        s_cum[i0]     += off;
        s_cum[i0 + 1] += off;
    }
    __syncthreads();

    // publish cumsum
    cum[b * ENC_T + tid * 2]     = s_cum[tid * 2];
    cum[b * ENC_T + tid * 2 + 1] = s_cum[tid * 2 + 1];

    // e_of_t via binary search: smallest e with cum[e] > t  (lo==ENC_T -> none)
    for (int i = 0; i < DEC_T / 256; ++i) {
        const int t = tid + i * 256;
        int lo = 0, hi = ENC_T;
        while (lo < hi) {
            const int mid = (lo + hi) >> 1;
            if (s_cum[mid] > t) hi = mid; else lo = mid + 1;
        }
        e_of_t[b * DEC_T + t] = (lo < ENC_T) ? lo : -1;
    }
}

// ---------------------------------------------------------------------------
// Kernel 2: one block per (b, 16-row tile of t).
//   out[b, t0:t0+16, :]  via V_WMMA_F32_16X16X4_F32 over the <=16-wide one-hot
//   K window (exact f32: A entries are 0.0/1.0, so this reproduces the f32
//   gather bit-exactly).
//   att[b, t0:t0+16, :]  one-hot tile via float4 streaming stores.
// 256 threads = 8 waves; each wave owns 64 output columns (4 N-tiles).
// All output stores are non-temporal (TH=NT): 128 MB of write-once data must
// not evict the reusable 16 MB encoder working set from L2.
// ---------------------------------------------------------------------------
__global__ __launch_bounds__(256)
void onehot_attend_wmma(const float* __restrict__ enc,     // [B][ENC_T][ENC_DIM]
                        const int*   __restrict__ cum,     // [B][ENC_T]
                        const int*   __restrict__ e_of_t,  // [B][DEC_T]
                        float* __restrict__ out,           // [B][DEC_T][ENC_DIM]
                        float* __restrict__ att)           // [B][DEC_T][ENC_T]
{
    const int blk  = blockIdx.x;
    const int b    = blk >> 7;           // / (DEC_T/16)
    const int t0   = (blk & 127) << 4;   // tile start in t
    const int tid  = threadIdx.x;
    const int lane = tid & 31;
    const int wave = tid >> 5;
    const int hi   = lane >> 4;          // half-wave select
    const int lm   = lane & 15;

    const int* cumb = cum + b * ENC_T;

    // Window base phoneme. If the whole tile is past the total duration,
    // clamp to the last phoneme: its end <= t0 so A evaluates to all zeros
    // and WMMA naturally produces zero rows (uniform code path, no branch).
    int e0 = e_of_t[b * DEC_T + t0];
    if (e0 < 0) e0 = ENC_T - 1;

    // Build A = one-hot alignment window (16 rows of t x 16 phonemes),
    // as 4 K-chunks in the f32 16x4 A layout:
    //   lane m (m = lane&15) holds row M=m; VGPR0 = K = 2*hi, VGPR1 = K = 2*hi+1
    v2f A[4];
    const int trow = t0 + lm;
    #pragma unroll
    for (int c = 0; c < 4; ++c) {
        float a0 = 0.0f, a1 = 0.0f;
        const int k0 = c * 4 + 2 * hi;
        const int ea = e0 + k0;
        const int eb = ea + 1;
        if (ea < ENC_T) {
            const int s  = (ea == 0) ? 0 : cumb[ea - 1];
            const int en = cumb[ea];
            a0 = (trow >= s && trow < en) ? 1.0f : 0.0f;
        }
        if (eb < ENC_T) {
            const int s  = cumb[eb - 1];
            const int en = cumb[eb];
            a1 = (trow >= s && trow < en) ? 1.0f : 0.0f;
        }
        A[c].x = a0;
        A[c].y = a1;
    }

    const float* encb = enc + (size_t)b * ENC_T * ENC_DIM;
    float*       outb = out + ((size_t)b * DEC_T + t0) * ENC_DIM;

    // 4 N-tiles of 16 columns per wave -> 64 columns; 8 waves cover 512.
    #pragma unroll
    for (int nt = 0; nt < 4; ++nt) {
        const int n0 = wave * 64 + nt * 16;
        v8f acc = {};
        #pragma unroll
        for (int c = 0; c < 4; ++c) {
            // B (4x16 K x N), mirror layout: lane n holds col N=n;
            // VGPR0 = row K=2*hi, VGPR1 = row K=2*hi+1 (clamped in-bounds;
            // values where A's column is zero are don't-cares).
            int r0 = e0 + c * 4 + 2 * hi;
            int r1 = r0 + 1;
            r0 = (r0 < ENC_T) ? r0 : (ENC_T - 1);
            r1 = (r1 < ENC_T) ? r1 : (ENC_T - 1);
            v2f Bm;
            Bm.x = encb[(size_t)r0 * ENC_DIM + n0 + lm];
            Bm.y = encb[(size_t)r1 * ENC_DIM + n0 + lm];
            // D = A*B + C, exact f32  -> v_wmma_f32_16x16x4_f32
            acc = __builtin_amdgcn_wmma_f32_16x16x4_f32(
                      false, A[c], false, Bm, (short)0, acc, false, false);
        }
        // D layout (16x16 f32): VGPR r -> row (r + 8*hi), col = n0 + lm.
        // Non-temporal: write-once streaming output.
        #pragma unroll
        for (int r = 0; r < 8; ++r)
            __builtin_nontemporal_store(acc[r],
                outb + (size_t)(r + 8 * hi) * ENC_DIM + n0 + lm);
    }

    // att tile: 16 rows x 512 cols, one-hot by comparing column vs e_of_t.
    float*     attb = att + ((size_t)b * DEC_T + t0) * ENC_T;
    const int* et   = e_of_t + b * DEC_T + t0;
    #pragma unroll
    for (int i = 0; i < (16 * ENC_T / 4) / 256; ++i) {   // 8 float4 per thread
        const int idx = tid + i * 256;     // 0..2047
        const int row = idx >> 7;          // 0..15
        const int c0  = (idx & 127) * 4;   // column group
        const int ev  = et[row];
        v4f v;
        v.x = (ev == c0 + 0) ? 1.0f : 0.0f;
        v.y = (ev == c0 + 1) ? 1.0f : 0.0f;
        v.z = (ev == c0 + 2) ? 1.0f : 0.0f;
        v.w = (ev == c0 + 3) ? 1.0f : 0.0f;
        __builtin_nontemporal_store(v, (v4f*)(attb + (size_t)row * ENC_T + c0));
    }
}

// ---------------------------------------------------------------------------
// Inputs (setup_inputs order):
//   d_in[0] encoder_outputs f32 [B, ENC_T, ENC_DIM]
//   d_in[1] melenc_outputs  f32 (unused by reference math)
//   d_in[2] encoder_lengths f32 [B, ENC_T]
//   d_in[3] output_lengths  i32 (all == DEC_T, unused)
//   d_in[4] dec_T scalar (compile-time constant here)
// Output: out [B,DEC_T,ENC_DIM] f32  ++  att [B,DEC_T,ENC_T] f32
// ---------------------------------------------------------------------------
extern "C" void kernel_launch(void* const* d_in, const int* in_sizes, int n_in,
                              void* d_out, int out_size, void* d_ws, size_t ws_size,
                              hipStream_t stream) {
    (void)in_sizes; (void)n_in; (void)out_size; (void)ws_size;

    const float* enc     = (const float*)d_in[0];
    const float* enc_len = (const float*)d_in[2];

    float* out = (float*)d_out;
    float* att = out + (size_t)B_DIM * DEC_T * ENC_DIM;

    int* cum    = (int*)d_ws;                 // B*ENC_T ints  (32 KB)
    int* e_of_t = cum + B_DIM * ENC_T;        // B*DEC_T ints (128 KB)

    build_alignment_map<<<B_DIM, 256, 0, stream>>>(enc_len, cum, e_of_t);
    onehot_attend_wmma<<<B_DIM * (DEC_T / 16), 256, 0, stream>>>(
        enc, cum, e_of_t, out, att);
}